// HyperbolicFusion_83708912599139
// MI455X (gfx1250) — compile-verified
//
#include <hip/hip_runtime.h>
#include <hip/hip_bf16.h>
#include <math.h>

// Problem shape (fixed by setup_inputs)
#define B_Q      2048
#define SEQ      768
#define D        128
#define N_NODES  100000
#define TOPK     5
#define QG       128          // query groups of 16 (2048/16)
#define NCHUNK   50           // node chunks
#define TILES_PER_CHUNK 125   // 125 * 16 = 2000 nodes per chunk
#define LROW     132          // padded LDS row stride in floats (528 B = 33 * 16 B, 132 % 64 == 4)

typedef __attribute__((ext_vector_type(2))) float v2f;
typedef __attribute__((ext_vector_type(8))) float v8f;
typedef int v4i __attribute__((vector_size(16)));   // 128-bit chunk for async-to-LDS builtin

#if defined(__has_builtin)
#  if __has_builtin(__builtin_amdgcn_global_load_async_to_lds_b128)
#    define HAVE_ASYNC_LDS 1
#  endif
#endif
#ifndef HAVE_ASYNC_LDS
#  define HAVE_ASYNC_LDS 0
#endif

__device__ __forceinline__ void wait_async_all() {
#if HAVE_ASYNC_LDS
#  if defined(__has_builtin) && __has_builtin(__builtin_amdgcn_s_wait_asynccnt)
  __builtin_amdgcn_s_wait_asynccnt(0);
#  else
  asm volatile("s_wait_asynccnt 0" ::: "memory");
#  endif
#endif
}

// Copy a 16x128 f32 tile (8 KB) from global into LDS with row stride LROW.
// 512 16-byte chunks; lane handles chunks {lane + 32*j}. Async path uses the
// CDNA5 async-to-LDS instructions (ASYNCcnt tracked), fallback is plain copies.
// LDS row stride is 528 bytes == 33 v4i units, so v4i indexing stays exact.
__device__ __forceinline__ void load_tile16x128(const float* __restrict__ g,
                                                float* lds_base, int lane) {
#if HAVE_ASYNC_LDS
  v4i* gp = (v4i*)g;          // generic ptr into global; implicit cast to __device__ int4*
  v4i* lp = (v4i*)lds_base;   // generic ptr into LDS;    implicit cast to __shared__ int4*
#pragma unroll
  for (int j = 0; j < 16; ++j) {
    int c    = lane + 32 * j;   // chunk id 0..511
    int row  = c >> 5;          // 0..15
    int col4 = c & 31;          // float4 within row
    __builtin_amdgcn_global_load_async_to_lds_b128(gp + c, lp + row * 33 + col4, 0, 0);
  }
#else
#pragma unroll
  for (int j = 0; j < 16; ++j) {
    int c    = lane + 32 * j;
    int row  = c >> 5;
    int col4 = c & 31;
    float4 v = ((const float4*)g)[c];
    *(float4*)(lds_base + row * LROW + col4 * 4) = v;
  }
#endif
}

__device__ __forceinline__ void topk5_insert(float d, int i, float bd[TOPK], int bi[TOPK]) {
  if (d < bd[TOPK - 1]) {
    bd[TOPK - 1] = d; bi[TOPK - 1] = i;
#pragma unroll
    for (int j = TOPK - 1; j > 0; --j) {
      if (bd[j] < bd[j - 1]) {
        float td = bd[j]; bd[j] = bd[j - 1]; bd[j - 1] = td;
        int   ti = bi[j]; bi[j] = bi[j - 1]; bi[j - 1] = ti;
      }
    }
  }
}

// ---------------------------------------------------------------------------
// K0: z_proj = z_seq @ W.T + b ; z_hyp = expmap0 ; zs = -2 * logmap0(z_hyp)
// One block per query row, 128 threads (one per output dim).
// ---------------------------------------------------------------------------
__global__ __launch_bounds__(128) void proj_expmap_kernel(
    const float* __restrict__ z_seq, const float* __restrict__ W,
    const float* __restrict__ bias, float* __restrict__ z_hyp_out,
    float* __restrict__ zs_out) {
  const int row = blockIdx.x;
  const int d   = threadIdx.x;
  __shared__ float zrow[SEQ];
  __shared__ float red[D];

  for (int i = d; i < SEQ; i += D) zrow[i] = z_seq[(size_t)row * SEQ + i];
  __syncthreads();

  const float* w = W + (size_t)d * SEQ;
  float acc = bias[d];
#pragma unroll 4
  for (int k = 0; k < SEQ; k += 4) {
    float4 wv = *(const float4*)(w + k);
    acc += wv.x * zrow[k] + wv.y * zrow[k + 1] + wv.z * zrow[k + 2] + wv.w * zrow[k + 3];
  }

  red[d] = acc * acc;
  __syncthreads();
  for (int s = D / 2; s > 0; s >>= 1) {
    if (d < s) red[d] += red[d + s];
    __syncthreads();
  }
  float n2 = red[0];

  float n   = fmaxf(sqrtf(n2), 1e-7f);        // clip(||u||, EPS)
  float th  = tanhf(n);
  float hyp = th / n * acc;                   // z_hyp element
  float nh  = fmaxf(th, 1e-7f);               // clip(||z_hyp||, EPS) == tanh(n)
  float arg = fminf(fmaxf(nh, 1e-7f), 1.0f - 1e-5f);
  float at  = 0.5f * logf((1.0f + arg) / (1.0f - arg));   // artanh
  float ztan = at / nh * hyp;

  z_hyp_out[(size_t)row * D + d] = hyp;
  zs_out[(size_t)row * D + d]    = -2.0f * ztan;          // fold -2 for the score GEMM
}

// ---------------------------------------------------------------------------
// K1: per-node logmap0 scale s(r) = artanh(clip(||emb||))/||emb||, and
// nsq(r) = s^2 * ||emb||^2 = ||n_tan||^2. One wave per node row.
// ---------------------------------------------------------------------------
__global__ __launch_bounds__(256) void node_scale_kernel(
    const float* __restrict__ emb, float* __restrict__ s_node,
    float* __restrict__ nsq, int n) {
  const int lane = threadIdx.x & 31;
  const int r    = blockIdx.x * 8 + (threadIdx.x >> 5);
  if (r >= n) return;
  const float4 v = ((const float4*)(emb + (size_t)r * D))[lane];
  float ss = v.x * v.x + v.y * v.y + v.z * v.z + v.w * v.w;
#pragma unroll
  for (int off = 16; off > 0; off >>= 1) ss += __shfl_xor(ss, off, 32);
  float nn  = fmaxf(sqrtf(ss), 1e-7f);
  float arg = fminf(fmaxf(nn, 1e-7f), 1.0f - 1e-5f);
  float at  = 0.5f * logf((1.0f + arg) / (1.0f - arg));
  float s   = at / nn;
  if (lane == 0) { s_node[r] = s; nsq[r] = s * s * ss; }
}

// ---------------------------------------------------------------------------
// K2: fused score GEMM + per-lane top-5.
// score(node, query) = ||n_tan||^2 - 2 * z_tan . n_tan  (monotone in d2)
// Computed transposed (rows = nodes, cols = queries) with V_WMMA_F32_16X16X4_F32
// so each lane privately sees 8 node candidates per tile for ONE query.
// One wave per block: 16 queries x 2000-node chunk, double-buffered async LDS.
// ---------------------------------------------------------------------------
__global__ __launch_bounds__(32) void dist_topk_kernel(
    const float* __restrict__ node_emb, const float* __restrict__ zs,
    const float* __restrict__ s_node, const float* __restrict__ nsq,
    float* __restrict__ cand_d, int* __restrict__ cand_i) {
  __shared__ float zsh[16 * LROW];
  __shared__ float ntile[2][16 * LROW];

  const int lane  = threadIdx.x;
  const int qg    = blockIdx.x & (QG - 1);
  const int chunk = blockIdx.x >> 7;          // QG == 128
  const int q0    = qg * 16;
  const int node0 = chunk * (TILES_PER_CHUNK * 16);

  const int  q    = lane & 15;                // query column for this lane
  const bool hi   = lane >= 16;
  const int  ko   = hi ? 2 : 0;               // K sub-offset per ISA 16x4 layout
  const int  mrow = lane & 15;                // A-matrix row (node within tile)

  load_tile16x128(zs + (size_t)q0 * D, zsh, lane);
  load_tile16x128(node_emb + (size_t)node0 * D, ntile[0], lane);
  wait_async_all();
  __syncthreads();

  float bd[TOPK]; int bi[TOPK];
#pragma unroll
  for (int j = 0; j < TOPK; ++j) { bd[j] = 3.4e38f; bi[j] = 0; }

  for (int t = 0; t < TILES_PER_CHUNK; ++t) {
    const int buf = t & 1;
    if (t + 1 < TILES_PER_CHUNK)
      load_tile16x128(node_emb + (size_t)(node0 + (t + 1) * 16) * D, ntile[buf ^ 1], lane);

    const float  sm = s_node[node0 + t * 16 + mrow];  // n_tan = sm * emb
    const float* nt = &ntile[buf][mrow * LROW];
    const float* zq = &zsh[q * LROW];

    v8f c0 = {0.f, 0.f, 0.f, 0.f, 0.f, 0.f, 0.f, 0.f};
    v8f c1 = {0.f, 0.f, 0.f, 0.f, 0.f, 0.f, 0.f, 0.f};
#pragma unroll
    for (int kk = 0; kk < D; kk += 8) {       // two K=4 WMMAs per step, 2 accumulators
      v2f a, b;
      a[0] = nt[kk + ko] * sm;  a[1] = nt[kk + ko + 1] * sm;
      b[0] = zq[kk + ko];       b[1] = zq[kk + ko + 1];
      c0 = __builtin_amdgcn_wmma_f32_16x16x4_f32(false, a, false, b, (short)0, c0, false, false);
      a[0] = nt[kk + 4 + ko] * sm;  a[1] = nt[kk + 4 + ko + 1] * sm;
      b[0] = zq[kk + 4 + ko];       b[1] = zq[kk + 4 + ko + 1];
      c1 = __builtin_amdgcn_wmma_f32_16x16x4_f32(false, a, false, b, (short)0, c1, false, false);
    }

#pragma unroll
    for (int v = 0; v < 8; ++v) {
      const int m = node0 + t * 16 + v + (hi ? 8 : 0);  // C layout: vgpr v -> M = v (+8 hi half)
      const float score = c0[v] + c1[v] + nsq[m];
      topk5_insert(score, m, bd, bi);
    }

    wait_async_all();
    __syncthreads();   // single-wave WG: S_NOP, but orders LDS for fallback path
  }

  // merge hi-half lane (same query, other 8-node rows) into lo-half lane
#pragma unroll
  for (int j = 0; j < TOPK; ++j) {
    float od = __shfl_xor(bd[j], 16, 32);
    int   oi = __shfl_xor(bi[j], 16, 32);
    if (!hi) topk5_insert(od, oi, bd, bi);
  }

  if (lane < 16) {
    const size_t base = ((size_t)(q0 + lane) * NCHUNK + chunk) * TOPK;
#pragma unroll
    for (int j = 0; j < TOPK; ++j) { cand_d[base + j] = bd[j]; cand_i[base + j] = bi[j]; }
  }
}

// ---------------------------------------------------------------------------
// K3: merge chunk candidates -> global top-5, gather-mean node_emb, mobius_add.
// One block per query, 128 threads (one per dim).
// ---------------------------------------------------------------------------
__global__ __launch_bounds__(128) void fuse_kernel(
    const float* __restrict__ node_emb, const float* __restrict__ z_hyp,
    const float* __restrict__ cand_d, const int* __restrict__ cand_i,
    float* __restrict__ z_fused) {
  const int qid = blockIdx.x;
  const int d   = threadIdx.x;
  __shared__ float cd[NCHUNK * TOPK];
  __shared__ int   ci[NCHUNK * TOPK];
  __shared__ int   top[TOPK];
  __shared__ float r0[D], r1[D], r2[D];

  for (int i = d; i < NCHUNK * TOPK; i += D) {
    cd[i] = cand_d[(size_t)qid * NCHUNK * TOPK + i];
    ci[i] = cand_i[(size_t)qid * NCHUNK * TOPK + i];
  }
  __syncthreads();

  if (d == 0) {
    float bd[TOPK]; int bi[TOPK];
#pragma unroll
    for (int j = 0; j < TOPK; ++j) { bd[j] = 3.4e38f; bi[j] = 0; }
    for (int i = 0; i < NCHUNK * TOPK; ++i) topk5_insert(cd[i], ci[i], bd, bi);
#pragma unroll
    for (int j = 0; j < TOPK; ++j) top[j] = bi[j];
  }
  __syncthreads();

  float nearv = 0.f;
#pragma unroll
  for (int j = 0; j < TOPK; ++j) nearv += node_emb[(size_t)top[j] * D + d];
  nearv *= (1.0f / TOPK);

  const float x = z_hyp[(size_t)qid * D + d];
  r0[d] = x * x; r1[d] = nearv * nearv; r2[d] = x * nearv;
  __syncthreads();
  for (int s = D / 2; s > 0; s >>= 1) {
    if (d < s) { r0[d] += r0[d + s]; r1[d] += r1[d + s]; r2[d] += r2[d + s]; }
    __syncthreads();
  }
  const float x2 = r0[0], y2 = r1[0], xy = r2[0];
  const float num = (1.0f + 2.0f * xy + y2) * x + (1.0f - x2) * nearv;
  const float den = 1.0f + 2.0f * xy + x2 * y2;
  z_fused[(size_t)qid * D + d] = num / (den + 1e-7f);
}

// ---------------------------------------------------------------------------
extern "C" void kernel_launch(void* const* d_in, const int* in_sizes, int n_in,
                              void* d_out, int out_size, void* d_ws, size_t ws_size,
                              hipStream_t stream) {
  (void)in_sizes; (void)n_in; (void)out_size; (void)ws_size;
  const float* z_seq    = (const float*)d_in[0];
  const float* node_emb = (const float*)d_in[1];
  const float* W        = (const float*)d_in[2];
  const float* bias     = (const float*)d_in[3];

  float* out     = (float*)d_out;
  float* z_fused = out;                      // (B, D)
  float* z_hyp   = out + (size_t)B_Q * D;    // (B, D)

  float* ws     = (float*)d_ws;
  float* zs     = ws;                                   // B*D        = 262144
  float* s_node = zs + (size_t)B_Q * D;                 // N          = 100000
  float* nsq    = s_node + N_NODES;                     // N          = 100000
  float* cand_d = nsq + N_NODES;                        // B*NCHUNK*5 = 512000
  int*   cand_i = (int*)(cand_d + (size_t)B_Q * NCHUNK * TOPK);

  proj_expmap_kernel<<<B_Q, D, 0, stream>>>(z_seq, W, bias, z_hyp, zs);
  node_scale_kernel<<<(N_NODES + 7) / 8, 256, 0, stream>>>(node_emb, s_node, nsq, N_NODES);
  dist_topk_kernel<<<QG * NCHUNK, 32, 0, stream>>>(node_emb, zs, s_node, nsq, cand_d, cand_i);
  fuse_kernel<<<B_Q, D, 0, stream>>>(node_emb, z_hyp, cand_d, cand_i, z_fused);
}